// EfficientCausalAttention_50019189129348
// MI455X (gfx1250) — compile-verified
//
#include <hip/hip_runtime.h>
#include <hip/hip_bf16.h>

typedef __attribute__((ext_vector_type(2))) float v2f;
typedef __attribute__((ext_vector_type(8))) float v8f;

// D = A(16x4) * B(4x16) + C, fp32, wave32
#define WMMA_F32(a, b, c) \
    __builtin_amdgcn_wmma_f32_16x16x4_f32(false, (a), false, (b), (short)0, (c), false, false)

namespace {
constexpr int kB   = 4096;
constexpr int kN   = 32;     // tokens
constexpr int kH   = 8;      // heads
constexpr int kD   = 64;     // head dim
constexpr int kHID = 512;    // kH * kD
constexpr int kQS  = 1540;   // sQKV row stride in floats (1536 + 4 pad; 1540 % 64 == 4)
constexpr int kPS  = 36;     // prob row stride (32 + 4 pad)
}

__global__ __launch_bounds__(256)
void attn_fused_kernel(const float* __restrict__ X,
                       const unsigned char* __restrict__ mask,
                       const float* __restrict__ Wqkv,
                       const float* __restrict__ bqkv,
                       const float* __restrict__ Wo,
                       const float* __restrict__ bo,
                       float* __restrict__ out,
                       float* __restrict__ attn_out)
{
    __shared__ float sQKV[kN * kQS];        // [32][1540] : q|k|v columns 0..1535
    __shared__ float sProb[kH * kN * kPS];  // per-head scores/probs
    __shared__ float sMask[kN * kN];        // additive mask bias (0 or -1e9)

    const int b    = blockIdx.x;
    const int tid  = threadIdx.x;
    const int wid  = tid >> 5;    // wave id 0..7
    const int lane = tid & 31;
    const int l16  = lane & 15;
    const int lh   = lane >> 4;   // 0 (K even pair) / 1 (K odd pair)

    // ---- mask -> additive bias in LDS ----
    for (int i = tid; i < kN * kN; i += 256)
        sMask[i] = mask[i] ? 0.0f : -1.0e9f;

    const float* __restrict__ Xb = X + (size_t)b * kN * kHID;

    // ================= Phase 1: qkv = Xb @ Wqkv^T + bqkv -> sQKV ==========
    // 96 output col-tiles of 16; each wave does 2 col-tiles x 2 row-tiles per pass.
    for (int pass = 0; pass < 6; ++pass) {
        const int nt0 = pass * 16 + wid * 2;
        v8f acc[2][2] = {};
        for (int kk = 0; kk < kHID; kk += 4) {
            v2f a0 = *(const v2f*)(Xb + (size_t)(l16     ) * kHID + kk + 2 * lh);
            v2f a1 = *(const v2f*)(Xb + (size_t)(l16 + 16) * kHID + kk + 2 * lh);
            v2f b0 = *(const v2f*)(Wqkv + ((size_t)(nt0    ) * 16 + l16) * kHID + kk + 2 * lh);
            v2f b1 = *(const v2f*)(Wqkv + ((size_t)(nt0 + 1) * 16 + l16) * kHID + kk + 2 * lh);
            acc[0][0] = WMMA_F32(a0, b0, acc[0][0]);
            acc[1][0] = WMMA_F32(a1, b0, acc[1][0]);
            acc[0][1] = WMMA_F32(a0, b1, acc[0][1]);
            acc[1][1] = WMMA_F32(a1, b1, acc[1][1]);
        }
        #pragma unroll
        for (int j = 0; j < 2; ++j) {
            const int col   = (nt0 + j) * 16 + l16;
            const float bia = bqkv[col];
            #pragma unroll
            for (int mt = 0; mt < 2; ++mt)
                #pragma unroll
                for (int g = 0; g < 8; ++g) {
                    const int row = mt * 16 + g + 8 * lh;   // C layout: v[g] = M=g (lo lanes) / g+8 (hi)
                    sQKV[row * kQS + col] = acc[mt][j][g] + bia;
                }
        }
    }
    __syncthreads();

    // ================= Phase 2a: scores = q k^T * 1/8 + mask ==============
    {
        const int h  = wid;               // one head per wave
        const int qo = h * kD;
        const int ko = kHID + h * kD;
        float* __restrict__ prob = sProb + h * kN * kPS;

        v8f sc[2][2] = {};
        for (int kk = 0; kk < kD; kk += 4) {
            v2f a0 = *(const v2f*)(&sQKV[(l16     ) * kQS + qo + kk + 2 * lh]);
            v2f a1 = *(const v2f*)(&sQKV[(l16 + 16) * kQS + qo + kk + 2 * lh]);
            v2f b0 = *(const v2f*)(&sQKV[(l16     ) * kQS + ko + kk + 2 * lh]);  // kT cols 0..15
            v2f b1 = *(const v2f*)(&sQKV[(l16 + 16) * kQS + ko + kk + 2 * lh]);  // kT cols 16..31
            sc[0][0] = WMMA_F32(a0, b0, sc[0][0]);
            sc[1][0] = WMMA_F32(a1, b0, sc[1][0]);
            sc[0][1] = WMMA_F32(a0, b1, sc[0][1]);
            sc[1][1] = WMMA_F32(a1, b1, sc[1][1]);
        }
        const float inv_scale = 0.125f;   // 1/sqrt(64)
        #pragma unroll
        for (int mt = 0; mt < 2; ++mt)
            #pragma unroll
            for (int jt = 0; jt < 2; ++jt)
                #pragma unroll
                for (int g = 0; g < 8; ++g) {
                    const int row = mt * 16 + g + 8 * lh;
                    const int col = jt * 16 + l16;
                    prob[row * kPS + col] = sc[mt][jt][g] * inv_scale + sMask[row * kN + col];
                }
    }
    __syncthreads();

    // ================= Phase 2b: softmax (lane = row), emit attn ==========
    {
        float* __restrict__ prob = sProb + wid * kN * kPS;
        const int r = lane;
        float v[kN];
        float mx = -3.0e38f;
        #pragma unroll
        for (int m = 0; m < kN; ++m) { v[m] = prob[r * kPS + m]; mx = fmaxf(mx, v[m]); }
        float sum = 0.0f;
        #pragma unroll
        for (int m = 0; m < kN; ++m) { v[m] = __expf(v[m] - mx); sum += v[m]; }
        const float inv = 1.0f / sum;
        float* __restrict__ arow =
            attn_out + (((size_t)b * kH + wid) * kN + r) * kN;
        #pragma unroll
        for (int m = 0; m < kN; ++m) {
            const float p = v[m] * inv;
            prob[r * kPS + m] = p;
            arow[m] = p;
        }
    }
    __syncthreads();

    // ================= Phase 2c: ctx = probs @ v (overwrite q cols) =======
    {
        const int h  = wid;
        const int qo = h * kD;
        const int vo = 2 * kHID + h * kD;
        float* __restrict__ prob = sProb + h * kN * kPS;

        for (int dt = 0; dt < 4; ++dt) {
            v8f acc[2] = {};
            for (int kk = 0; kk < kN; kk += 4) {
                v2f a0 = *(const v2f*)(&prob[(l16     ) * kPS + kk + 2 * lh]);
                v2f a1 = *(const v2f*)(&prob[(l16 + 16) * kPS + kk + 2 * lh]);
                v2f bb;
                bb.x = sQKV[(kk + 2 * lh    ) * kQS + vo + dt * 16 + l16];  // B row K, col N
                bb.y = sQKV[(kk + 2 * lh + 1) * kQS + vo + dt * 16 + l16];
                acc[0] = WMMA_F32(a0, bb, acc[0]);
                acc[1] = WMMA_F32(a1, bb, acc[1]);
            }
            #pragma unroll
            for (int mt = 0; mt < 2; ++mt)
                #pragma unroll
                for (int g = 0; g < 8; ++g) {
                    const int row = mt * 16 + g + 8 * lh;
                    sQKV[row * kQS + qo + dt * 16 + l16] = acc[mt][g];
                }
        }
    }
    __syncthreads();

    // ================= Phase 3: out = ctx @ Wo^T + bo =====================
    float* __restrict__ outb = out + (size_t)b * kN * kHID;
    for (int pass = 0; pass < 2; ++pass) {
        const int nt0 = pass * 16 + wid * 2;
        v8f acc[2][2] = {};
        for (int kk = 0; kk < kHID; kk += 4) {
            v2f a0 = *(const v2f*)(&sQKV[(l16     ) * kQS + kk + 2 * lh]);
            v2f a1 = *(const v2f*)(&sQKV[(l16 + 16) * kQS + kk + 2 * lh]);
            v2f b0 = *(const v2f*)(Wo + ((size_t)(nt0    ) * 16 + l16) * kHID + kk + 2 * lh);
            v2f b1 = *(const v2f*)(Wo + ((size_t)(nt0 + 1) * 16 + l16) * kHID + kk + 2 * lh);
            acc[0][0] = WMMA_F32(a0, b0, acc[0][0]);
            acc[1][0] = WMMA_F32(a1, b0, acc[1][0]);
            acc[0][1] = WMMA_F32(a0, b1, acc[0][1]);
            acc[1][1] = WMMA_F32(a1, b1, acc[1][1]);
        }
        #pragma unroll
        for (int j = 0; j < 2; ++j) {
            const int col   = (nt0 + j) * 16 + l16;
            const float bia = bo[col];
            #pragma unroll
            for (int mt = 0; mt < 2; ++mt)
                #pragma unroll
                for (int g = 0; g < 8; ++g) {
                    const int row = mt * 16 + g + 8 * lh;
                    outb[(size_t)row * kHID + col] = acc[mt][j][g] + bia;
                }
        }
    }
}

extern "C" void kernel_launch(void* const* d_in, const int* in_sizes, int n_in,
                              void* d_out, int out_size, void* d_ws, size_t ws_size,
                              hipStream_t stream) {
    const float*         X    = (const float*)d_in[0];
    const unsigned char* mask = (const unsigned char*)d_in[1];  // jnp bool -> 1 byte
    const float*         Wqkv = (const float*)d_in[2];
    const float*         bqkv = (const float*)d_in[3];
    const float*         Wo   = (const float*)d_in[4];
    const float*         bo   = (const float*)d_in[5];

    float* out  = (float*)d_out;                          // [4096,32,512]
    float* attn = out + (size_t)kB * kN * kHID;           // [4096,8,32,32]

    hipLaunchKernelGGL(attn_fused_kernel, dim3(kB), dim3(256), 0, stream,
                       X, mask, Wqkv, bqkv, Wo, bo, out, attn);
}